// EnKoMa_80221399155263
// MI455X (gfx1250) — compile-verified
//
#include <hip/hip_runtime.h>
#include <hip/hip_bf16.h>

// ---------------------------------------------------------------------------
// Problem constants (match reference setup_inputs)
// ---------------------------------------------------------------------------
#define BZ     8
#define TW     1024
#define BT     (BZ * TW)        // 8192 rows
#define DIN    32
#define DM     64               // d_model
#define DI     128              // d_inner
#define DTRK   4                // dt_rank
#define DST    32               // D_STATE
#define XD     (DTRK + 2*DST)   // 68 = logical xproj output width
#define XDP    80               // padded xproj width (multiple of 16)
#define CURRT  (TW + 4)         // curr buffer time length
#define PRED   4

typedef float v2f __attribute__((ext_vector_type(2)));
typedef float v8f __attribute__((ext_vector_type(8)));

#if defined(__has_builtin)
#  if __has_builtin(__builtin_amdgcn_wmma_f32_16x16x4_f32)
#    define HAVE_WMMA_F32 1
#  else
#    define HAVE_WMMA_F32 0
#  endif
#else
#  define HAVE_WMMA_F32 0
#endif

// ---------------------------------------------------------------------------
// fp32 WMMA GEMM:  out[M,N] (=|+=) act(A[M,K] @ W[K,N] + bias)
// One wave per 16x16 output tile; V_WMMA_F32_16X16X4_F32, K consumed 4/inst.
// N is always a multiple of 16 (ragged weights pre-padded with zeros) and
// RELU/ACCUM/HASBIAS are template params, so both the K-loop and the epilogue
// are completely branch-free (EXEC stays full for every WMMA).
// Fragment layouts per CDNA5 ISA 7.12.2:
//   A (16x4):  lane<16 -> row M=lane, v0=K0,v1=K1 ; lane>=16 -> K2,K3
//   B (4x16):  lane<16 -> col N=lane, v0=K0,v1=K1 ; lane>=16 -> K2,K3
//   D (16x16): vgpr r  -> (M=r, N=lane) lanes 0-15 ; (M=r+8) lanes 16-31
// ---------------------------------------------------------------------------
template <int RELU, int ACCUM, int HASBIAS>
__global__ void __launch_bounds__(256)
enkoma_gemm_f32_wmma(const float* __restrict__ A, const float* __restrict__ W,
                     const float* __restrict__ bias, float* __restrict__ out,
                     int M, int K, int N, int nTN)
{
  const int wave = threadIdx.x >> 5;
  const int lane = threadIdx.x & 31;
  const int tile = blockIdx.x * 8 + wave;
  const int totalTiles = (M >> 4) * nTN;
  if (tile >= totalTiles) return;            // whole-wave exit: EXEC stays full
  const int tm = tile / nTN;
  const int tn = tile - tm * nTN;
  const int m0 = tm << 4, n0 = tn << 4;
  const int half = lane >> 4, lr = lane & 15;
  const int n = n0 + lr;

  v8f acc = {0.f, 0.f, 0.f, 0.f, 0.f, 0.f, 0.f, 0.f};

  const float* pA = A + (size_t)(m0 + lr) * K + 2 * half;   // K0/K1 or K2/K3
  const float* pB = W + (size_t)(2 * half) * N + n;

#if HAVE_WMMA_F32
#pragma unroll 4
  for (int k = 0; k < K; k += 4) {           // K is a multiple of 16
    v2f a, b;
    a.x = pA[0];
    a.y = pA[1];
    b.x = pB[0];
    b.y = pB[N];
    acc = __builtin_amdgcn_wmma_f32_16x16x4_f32(false, a, false, b,
                                                (short)0, acc, false, false);
    pA += 4;
    pB += (size_t)4 * N;
  }
#else
  for (int k = 0; k < K; k += 4) {
#pragma unroll
    for (int r = 0; r < 8; ++r) {
      const float* Ar = A + (size_t)(m0 + r + 8 * half) * K;
      float s = acc[r];
#pragma unroll
      for (int kk = 0; kk < 4; ++kk)
        s += Ar[k + kk] * W[(size_t)(k + kk) * N + n];
      acc[r] = s;
    }
  }
#endif

  const float bv = HASBIAS ? bias[n] : 0.f;
  float* po = out + (size_t)(m0 + 8 * half) * N + n;
#pragma unroll
  for (int r = 0; r < 8; ++r) {
    float v = acc[r] + bv;
    if (RELU) v = fmaxf(v, 0.f);
    if (ACCUM) *po += v; else *po = v;
    po += N;
  }
}

// zero-pad Wxproj (NB x 128 x 68) -> Wp (NB x 128 x 80)
__global__ void __launch_bounds__(256)
enkoma_pad_wxproj(const float* __restrict__ Wx, float* __restrict__ Wp)
{
  const int idx = blockIdx.x * 256 + threadIdx.x;
  if (idx >= 2 * DI * XDP) return;
  const int c = idx % XDP;
  const int row = idx / XDP;             // blk*128 + k
  Wp[idx] = (c < XD) ? Wx[row * XD + c] : 0.f;
}

// ---------------------------------------------------------------------------
// Depthwise causal conv (K=4) over time + bias + SiLU.
// xz rows are 256 wide: cols [0,128)=xi-preconv, [128,256)=res.
// ---------------------------------------------------------------------------
__global__ void __launch_bounds__(256)
enkoma_conv_silu(const float* __restrict__ xz, const float* __restrict__ Wc,
                 const float* __restrict__ bc, float* __restrict__ xi)
{
  const int idx = blockIdx.x * 256 + threadIdx.x;
  if (idx >= BT * DI) return;
  const int m = idx >> 7;          // row = b*TW + t
  const int d = idx & (DI - 1);
  const int t = m & (TW - 1);
  float s = bc[d];
#pragma unroll
  for (int k = 0; k < 4; ++k) {
    const int tt = t - 3 + k;      // tap k reads x[t-3+k]
    if (tt >= 0) s += xz[(size_t)(m - 3 + k) * 256 + d] * Wc[d * 4 + k];
  }
  xi[idx] = s * (1.f / (1.f + __expf(-s)));   // SiLU
}

// ---------------------------------------------------------------------------
// dt = softplus(dtr @ Wdt + bdt)   (K=4 contraction, trivial)
// xdbl rows 80 wide (padded): [0,4)=dtr, [4,36)=B, [36,68)=C, [68,80)=pad
// ---------------------------------------------------------------------------
__global__ void __launch_bounds__(256)
enkoma_dt(const float* __restrict__ xdbl, const float* __restrict__ Wdt,
          const float* __restrict__ bdt, float* __restrict__ dt)
{
  const int idx = blockIdx.x * 256 + threadIdx.x;
  if (idx >= BT * DI) return;
  const int m = idx >> 7;
  const int d = idx & (DI - 1);
  float s = bdt[d];
#pragma unroll
  for (int r = 0; r < DTRK; ++r)
    s += xdbl[(size_t)m * XDP + r] * Wdt[r * DI + d];
  dt[idx] = (s > 20.f) ? s : log1pf(__expf(s));  // softplus
}

// ---------------------------------------------------------------------------
// Selective scan. grid=(8 batches, 4 d-chunks), block=1024 (32 waves).
// wave -> channel d, lane -> state s.  h = exp(dt*A)*h + (dt*x)*B[s];
// y[b,t,d] = sum_s h*C[s] via wave shfl reduction.
// ---------------------------------------------------------------------------
__global__ void __launch_bounds__(1024)
enkoma_scan(const float* __restrict__ dt, const float* __restrict__ xi,
            const float* __restrict__ xdbl, const float* __restrict__ Alog,
            float* __restrict__ y)
{
  const int b = blockIdx.x;
  const int wave = threadIdx.x >> 5;
  const int s = threadIdx.x & 31;
  const int d = blockIdx.y * 32 + wave;
  const float Aneg = -__expf(Alog[d * DST + s]);   // A = -exp(Alog)
  float h = 0.f;
  const int base = b * TW;
  for (int t = 0; t < TW; ++t) {
    const int m = base + t;
    const float dtv = dt[(size_t)m * DI + d];                // wave-uniform
    const float xv  = xi[(size_t)m * DI + d];                // wave-uniform
    const float Bv  = xdbl[(size_t)m * XDP + DTRK + s];
    const float Cv  = xdbl[(size_t)m * XDP + DTRK + DST + s];
    h = __expf(dtv * Aneg) * h + (dtv * xv) * Bv;
    float yp = h * Cv;
#pragma unroll
    for (int off = 16; off; off >>= 1) yp += __shfl_xor(yp, off, 32);
    if (s == 0) y[(size_t)m * DI + d] = yp;
  }
}

// y = (y_scan + xi*Dp) * silu(res)
__global__ void __launch_bounds__(256)
enkoma_ygate(float* __restrict__ y, const float* __restrict__ xi,
             const float* __restrict__ xz, const float* __restrict__ Dp)
{
  const int idx = blockIdx.x * 256 + threadIdx.x;
  if (idx >= BT * DI) return;
  const int m = idx >> 7;
  const int d = idx & (DI - 1);
  const float r = xz[(size_t)m * 256 + 128 + d];
  const float sil = r * (1.f / (1.f + __expf(-r)));
  y[idx] = (y[idx] + xi[idx] * Dp[d]) * sil;
}

// strided (windowed) source -> contiguous u[BT, DM]
__global__ void __launch_bounds__(256)
enkoma_copy_window(const float* __restrict__ src, float* __restrict__ u,
                   int srcBatchStride, int toff)
{
  const int idx = blockIdx.x * 256 + threadIdx.x;
  if (idx >= BT * DM) return;
  const int m = idx >> 6;
  const int d = idx & (DM - 1);
  const int b = m >> 10;
  const int t = m & (TW - 1);
  u[idx] = src[(size_t)b * srcBatchStride + (size_t)(toff + t) * DM + d];
}

// contiguous z -> curr[:, 0:TW, :]
__global__ void __launch_bounds__(256)
enkoma_copy_to_curr(const float* __restrict__ z, float* __restrict__ curr)
{
  const int idx = blockIdx.x * 256 + threadIdx.x;
  if (idx >= BT * DM) return;
  const int m = idx >> 6;
  const int d = idx & (DM - 1);
  const int b = m >> 10;
  const int t = m & (TW - 1);
  curr[((size_t)b * CURRT + t) * DM + d] = z[idx];
}

// full LayerNorm over last dim 64; one wave per row (2 elems/lane)
__global__ void __launch_bounds__(256)
enkoma_layernorm(const float* __restrict__ u, const float* __restrict__ g,
                 const float* __restrict__ be, float* __restrict__ dst, int nRows)
{
  const int wave = threadIdx.x >> 5;
  const int lane = threadIdx.x & 31;
  const int m = blockIdx.x * 8 + wave;
  if (m >= nRows) return;
  const float v0 = u[(size_t)m * DM + lane];
  const float v1 = u[(size_t)m * DM + 32 + lane];
  float s = v0 + v1, q = v0 * v0 + v1 * v1;
#pragma unroll
  for (int off = 16; off; off >>= 1) {
    s += __shfl_xor(s, off, 32);
    q += __shfl_xor(q, off, 32);
  }
  const float mean = s * (1.f / 64.f);
  const float var  = q * (1.f / 64.f) - mean * mean;
  const float rstd = rsqrtf(var + 1e-5f);
  dst[(size_t)m * DM + lane]      = (v0 - mean) * rstd * g[lane]      + be[lane];
  dst[(size_t)m * DM + 32 + lane] = (v1 - mean) * rstd * g[32 + lane] + be[32 + lane];
}

// LayerNorm of the LAST timestep only; writes both curr append slot & preds
__global__ void __launch_bounds__(256)
enkoma_layernorm_last(const float* __restrict__ u, const float* __restrict__ g,
                      const float* __restrict__ be, float* __restrict__ curr,
                      float* __restrict__ preds, int p)
{
  const int b = threadIdx.x >> 5;      // 8 waves, one per batch
  const int lane = threadIdx.x & 31;
  if (b >= BZ) return;
  const int m = b * TW + (TW - 1);
  const float v0 = u[(size_t)m * DM + lane];
  const float v1 = u[(size_t)m * DM + 32 + lane];
  float s = v0 + v1, q = v0 * v0 + v1 * v1;
#pragma unroll
  for (int off = 16; off; off >>= 1) {
    s += __shfl_xor(s, off, 32);
    q += __shfl_xor(q, off, 32);
  }
  const float mean = s * (1.f / 64.f);
  const float var  = q * (1.f / 64.f) - mean * mean;
  const float rstd = rsqrtf(var + 1e-5f);
  const float o0 = (v0 - mean) * rstd * g[lane]      + be[lane];
  const float o1 = (v1 - mean) * rstd * g[32 + lane] + be[32 + lane];
  float* cr = curr + ((size_t)b * CURRT + TW + p) * DM;
  cr[lane] = o0; cr[32 + lane] = o1;
  float* pr = preds + ((size_t)b * PRED + p) * DM;
  pr[lane] = o0; pr[32 + lane] = o1;
}

// ---------------------------------------------------------------------------
// Host-side orchestration
// ---------------------------------------------------------------------------
static inline void launch_gemm(hipStream_t st, const float* A, const float* W,
                               const float* bias, float* out,
                               int M, int K, int N, int relu, int accum)
{
  const int nTN = N >> 4;                 // N is always a multiple of 16
  const int tiles = (M >> 4) * nTN;
  const int wgs = (tiles + 7) >> 3;
  if (bias) {
    if (relu)       enkoma_gemm_f32_wmma<1,0,1><<<wgs,256,0,st>>>(A,W,bias,out,M,K,N,nTN);
    else            enkoma_gemm_f32_wmma<0,0,1><<<wgs,256,0,st>>>(A,W,bias,out,M,K,N,nTN);
  } else {
    if (accum)      enkoma_gemm_f32_wmma<0,1,0><<<wgs,256,0,st>>>(A,W,bias,out,M,K,N,nTN);
    else            enkoma_gemm_f32_wmma<0,0,0><<<wgs,256,0,st>>>(A,W,bias,out,M,K,N,nTN);
  }
}

struct MambaWs { float *u, *xz, *xi, *xdbl, *dtb, *yb, *Wp; };

struct MambaWts {
  const float *Win, *Wconv, *bconv, *Wdt, *bdt, *Alog, *Dp, *Wout;
};

static void run_mamba_block(hipStream_t st, const MambaWs& w, const MambaWts& wt, int blk)
{
  const int EW = (BT * DI + 255) / 256;
  // xz = u @ Win  (M=8192, K=64, N=256)
  launch_gemm(st, w.u, wt.Win + (size_t)blk * DM * 2 * DI, nullptr, w.xz,
              BT, DM, 2 * DI, 0, 0);
  // depthwise conv + bias + silu
  enkoma_conv_silu<<<EW, 256, 0, st>>>(w.xz, wt.Wconv + (size_t)blk * DI * 4,
                                       wt.bconv + (size_t)blk * DI, w.xi);
  // xdbl = xi @ Wxproj_padded  (K=128, N=80; cols 68..79 are zero)
  launch_gemm(st, w.xi, w.Wp + (size_t)blk * DI * XDP, nullptr, w.xdbl,
              BT, DI, XDP, 0, 0);
  // dt = softplus(dtr @ Wdt + bdt)
  enkoma_dt<<<EW, 256, 0, st>>>(w.xdbl, wt.Wdt + (size_t)blk * DTRK * DI,
                                wt.bdt + (size_t)blk * DI, w.dtb);
  // selective scan
  dim3 sg(BZ, DI / 32);
  enkoma_scan<<<sg, 1024, 0, st>>>(w.dtb, w.xi, w.xdbl,
                                   wt.Alog + (size_t)blk * DI * DST, w.yb);
  // gate: y = (y + xi*Dp) * silu(res)
  enkoma_ygate<<<EW, 256, 0, st>>>(w.yb, w.xi, w.xz, wt.Dp + (size_t)blk * DI);
  // residual: u += y @ Wout  (K=128, N=64)
  launch_gemm(st, w.yb, wt.Wout + (size_t)blk * DI * DM, nullptr, w.u,
              BT, DI, DM, 0, 1);
}

extern "C" void kernel_launch(void* const* d_in, const int* in_sizes, int n_in,
                              void* d_out, int out_size, void* d_ws, size_t ws_size,
                              hipStream_t stream)
{
  (void)in_sizes; (void)n_in; (void)out_size; (void)ws_size;
  const float* x      = (const float*)d_in[0];
  const float* We1    = (const float*)d_in[1];
  const float* be1    = (const float*)d_in[2];
  const float* We2    = (const float*)d_in[3];
  const float* be2    = (const float*)d_in[4];
  const float* Wd1    = (const float*)d_in[5];
  const float* bd1    = (const float*)d_in[6];
  const float* Wd2    = (const float*)d_in[7];
  const float* bd2    = (const float*)d_in[8];
  MambaWts wt;
  wt.Win    = (const float*)d_in[9];
  wt.Wconv  = (const float*)d_in[10];
  wt.bconv  = (const float*)d_in[11];
  const float* Wxproj = (const float*)d_in[12];
  wt.Wdt    = (const float*)d_in[13];
  wt.bdt    = (const float*)d_in[14];
  wt.Alog   = (const float*)d_in[15];
  wt.Dp     = (const float*)d_in[16];
  wt.Wout   = (const float*)d_in[17];
  const float* gamma  = (const float*)d_in[18];
  const float* beta   = (const float*)d_in[19];
  // d_in[20] = pred_len (int). Fixed at 4 by the reference; cannot be read
  // host-side inside graph capture, so the 4 prediction passes are unrolled.

  float* out = (float*)d_out;
  // output tuple layout: x_rec, x_dyn, x_pred, z, z_dyn (flat, in order)
  float* x_rec  = out;                                   // 8*1024*32
  float* x_dyn  = out + (size_t)BT * DIN;                // 8*1024*32
  float* x_pred = out + (size_t)2 * BT * DIN;            // 8*4*32
  float* z      = out + (size_t)2 * BT * DIN + BZ * PRED * DIN;   // 8*1024*64
  float* z_dyn  = z + (size_t)BT * DM;                   // 8*1024*64

  // workspace carve-up (floats); total ~28 MB, resident in MI455X's 192MB L2
  float* ws = (float*)d_ws;
  MambaWs w;
  w.u     = ws;                               // 524288
  w.xz    = w.u    + (size_t)BT * DM;         // 2097152
  w.xi    = w.xz   + (size_t)BT * 2 * DI;     // 1048576
  w.xdbl  = w.xi   + (size_t)BT * DI;         // 655360 (80-wide padded)
  w.dtb   = w.xdbl + (size_t)BT * XDP;        // 1048576
  w.yb    = w.dtb  + (size_t)BT * DI;         // 1048576
  float* curr  = w.yb + (size_t)BT * DI;      // 8*1028*64
  float* preds = curr + (size_t)BZ * CURRT * DM; // 8*4*64
  w.Wp    = preds + (size_t)BZ * PRED * DM;   // 2*128*80 padded Wxproj
  float* h1    = w.xz;                        // MLP hidden reuses xz region

  const int CW = (BT * DM + 255) / 256;

  // ---- pad Wxproj once (deterministic each call) ----
  enkoma_pad_wxproj<<<(2 * DI * XDP + 255) / 256, 256, 0, stream>>>(Wxproj, w.Wp);

  // ---- encoder: z = relu(x@We1+be1) @ We2 + be2 ----
  launch_gemm(stream, x,  We1, be1, h1, BT, DIN, DM, 1, 0);
  launch_gemm(stream, h1, We2, be2, z,  BT, DM,  DM, 0, 0);

  // ---- x_rec = decoder(z) ----
  launch_gemm(stream, z,  Wd1, bd1, h1,    BT, DM, DM,  1, 0);
  launch_gemm(stream, h1, Wd2, bd2, x_rec, BT, DM, DIN, 0, 0);

  // ---- z_dyn = dynamics(z) ----
  enkoma_copy_window<<<CW, 256, 0, stream>>>(z, w.u, TW * DM, 0);
  run_mamba_block(stream, w, wt, 0);
  run_mamba_block(stream, w, wt, 1);
  enkoma_layernorm<<<(BT + 7) / 8, 256, 0, stream>>>(w.u, gamma, beta, z_dyn, BT);

  // ---- x_dyn = decoder(z_dyn) ----
  launch_gemm(stream, z_dyn, Wd1, bd1, h1,    BT, DM, DM,  1, 0);
  launch_gemm(stream, h1,    Wd2, bd2, x_dyn, BT, DM, DIN, 0, 0);

  // ---- autoregressive prediction: 4 full dynamics passes over sliding windows
  enkoma_copy_to_curr<<<CW, 256, 0, stream>>>(z, curr);
  for (int p = 0; p < PRED; ++p) {
    enkoma_copy_window<<<CW, 256, 0, stream>>>(curr, w.u, CURRT * DM, p);
    run_mamba_block(stream, w, wt, 0);
    run_mamba_block(stream, w, wt, 1);
    enkoma_layernorm_last<<<1, 256, 0, stream>>>(w.u, gamma, beta, curr, preds, p);
  }

  // ---- x_pred = decoder(preds)  (M = 8*4 = 32 rows) ----
  launch_gemm(stream, preds, Wd1, bd1, h1,     BZ * PRED, DM, DM,  1, 0);
  launch_gemm(stream, h1,    Wd2, bd2, x_pred, BZ * PRED, DM, DIN, 0, 0);
}